// GraphRecSys_43396349558975
// MI455X (gfx1250) — compile-verified
//
#include <hip/hip_runtime.h>
#include <math.h>

// ---------------------------------------------------------------------------
// GraphRecSys on MI455X (gfx1250): fp32 WMMA (V_WMMA_F32_16X16X4_F32) GEMMs,
// double-buffered LDS staging, TDM (tensor_load_to_lds, 6-arg clang-23 form)
// for the A tile, row-wise LDS kernels for top-k / softmax / matvecs.
// N=2048, DU=DF=512, HID=256, OUT=256, H=4, K=5.
// ---------------------------------------------------------------------------

#define NN_ 2048
#define KSEL 6          // k+1 = 6 picks per row
#define ALPHA_ 0.2f
#define NEG_ -9.0e15f

typedef __attribute__((ext_vector_type(2))) float v2f;
typedef __attribute__((ext_vector_type(4))) float v4f;
typedef __attribute__((ext_vector_type(8))) float v8f;
typedef __attribute__((ext_vector_type(4))) unsigned u32x4;
typedef __attribute__((ext_vector_type(8))) int     i32x8;
typedef __attribute__((ext_vector_type(4))) int     i32x4;

// Device-pass-only guards: __HIP_DEVICE_COMPILE__ is never set on the host
// pass, so AS(3) pointer tricks are not even parsed for x86.
#if defined(__HIP_DEVICE_COMPILE__) && __has_builtin(__builtin_amdgcn_tensor_load_to_lds)
#define HAVE_TDM 1
#else
#define HAVE_TDM 0
#endif

#if defined(__HIP_DEVICE_COMPILE__)
static __device__ __forceinline__ unsigned lds_off_u32(const void* p) {
    // generic -> LDS(AS3) addrspacecast; AS3 pointers are 32-bit on amdgcn
    return (unsigned)(unsigned long long)
           (__attribute__((address_space(3))) const char*)p;
}
static __device__ __forceinline__ void wait_tensorcnt0() {
#if __has_builtin(__builtin_amdgcn_s_wait_tensorcnt)
    __builtin_amdgcn_s_wait_tensorcnt(0);
#else
    asm volatile("s_wait_tensorcnt 0x0" ::: "memory");
#endif
}
#endif

// A tile LDS row stride (floats): 16 data + 4 pad -> conflict-free frag reads
#define AP 20
// B tile LDS row stride (floats): 256 data (k-pair interleaved) + 32 pad
#define BP 288

// ---------------------------------------------------------------------------
// WMMA fp32 GEMM: C[M,N] = epi(A[M,K] @ op(B)),  op(B)=B ([K,N]) or B^T ([N,K])
// Block tile 128x128, 256 threads = 8 waves (4 along M x 2 along N),
// wave tile 32x64 = 2x4 WMMA 16x16 accumulators, K double-buffered 16/stage.
// EPI: 0=none, 1=ELU, 2=bias+ReLU.  Output goes to C[row*ldc + coff + col].
// ---------------------------------------------------------------------------
template<bool TRANS_B, int EPI>
__global__ __launch_bounds__(256, 2)
void gemm_wmma_f32(const float* __restrict__ A, const float* __restrict__ B,
                   const float* __restrict__ bias, float* __restrict__ C,
                   int M, int N, int K, int ldc, int coff)
{
    __shared__ __align__(16) float As[2][128][AP];   // 2 x 10 KB
    __shared__ __align__(16) float BsP[2][8][BP];    // 2 x 9 KB, pair-interleaved

    const int tid  = threadIdx.x;
    const int lane = tid & 31;
    const int wave = tid >> 5;
    const int l16  = lane & 15;
    const int lh   = lane >> 4;          // lane half (0: lanes 0-15, 1: 16-31)
    const int wm   = (wave & 3) * 32;    // wave M offset in block tile
    const int wn   = (wave >> 2) * 64;   // wave N offset in block tile
    const int bM   = blockIdx.y * 128;
    const int bN   = blockIdx.x * 128;

    v8f acc[2][4];
#pragma unroll
    for (int i = 0; i < 2; ++i)
#pragma unroll
        for (int j = 0; j < 4; ++j) { v8f z = {}; acc[i][j] = z; }

    // ---------------- staging helpers ----------------
#if !HAVE_TDM
    float ra[8];
#endif
    float rb[8];

    // A tile [128 x 16] for K-offset k0 -> buffer `buf`
    auto issueA = [&](int k0, int buf) {
#if HAVE_TDM
        if (wave == 0) {
            // Tensor DMA descriptor (D#), 2D tile 16x128 of 4-byte elements,
            // LDS pad: +4 DWORDs every 16 DWORDs -> row stride AP floats.
            const unsigned lds = lds_off_u32(&As[buf][0][0]);
            const unsigned long long ga =
                (unsigned long long)(const void*)(A + (size_t)bM * K + k0);
            u32x4 g0;
            g0.x = 1u;                                   // count=1, user mode
            g0.y = lds;                                  // lds_addr
            g0.z = (unsigned)(ga & 0xFFFFFFFFu);         // global_addr[31:0]
            g0.w = (unsigned)((ga >> 32) & 0x01FFFFFFu)  // global_addr[56:32]
                 | (2u << 30);                           // type = 2 (image)
            i32x8 g1;
            g1[0] = (2 << 16)                            // data_size = 4B
                  | (1 << 20)                            // pad_enable
                  | (3 << 22)                            // pad_interval: 16 DW
                  | (3 << 25);                           // pad_amount: 4 DW
            g1[1] = (K & 0xFFFF) << 16;                  // tensor_dim0 lo
            g1[2] = ((K >> 16) & 0xFFFF) | ((M & 0xFFFF) << 16);   // d0 hi|d1 lo
            g1[3] = ((M >> 16) & 0xFFFF) | (16 << 16);   // d1 hi | tile_dim0=16
            g1[4] = 128;                                 // tile_dim1=128, tile_dim2=0
            g1[5] = K;                                   // tensor_dim0_stride lo
            g1[6] = 0;                                   // d0 stride hi | d1 stride lo
            g1[7] = 0;
            i32x4 z4 = {0, 0, 0, 0};
            i32x8 z8 = {0, 0, 0, 0, 0, 0, 0, 0};
            __builtin_amdgcn_tensor_load_to_lds(g0, g1, z4, z4, z8, 0);
        }
#else
        const int r  = tid >> 1;
        const int cq = (tid & 1) * 8;
        const float* src = A + (size_t)(bM + r) * K + k0 + cq;
        *(v4f*)&ra[0] = *(const v4f*)src;
        *(v4f*)&ra[4] = *(const v4f*)(src + 4);
        (void)buf;
#endif
    };
    auto commitA = [&](int buf) {
#if !HAVE_TDM
        const int r  = tid >> 1;
        const int cq = (tid & 1) * 8;
        *(v4f*)&As[buf][r][cq]     = *(v4f*)&ra[0];
        *(v4f*)&As[buf][r][cq + 4] = *(v4f*)&ra[4];
#else
        (void)buf;
#endif
    };
    auto waitA = [&]() {
#if HAVE_TDM
        if (wave == 0) wait_tensorcnt0();
#endif
    };

    // B tile for K-offset k0 -> regs
    auto loadB = [&](int k0) {
        if (!TRANS_B) {               // B[K][N]: 8 floats along n, fixed k row
            const int r  = tid >> 4;
            const int cq = (tid & 15) * 8;
            const float* src = B + (size_t)(k0 + r) * N + bN + cq;
            *(v4f*)&rb[0] = *(const v4f*)src;
            *(v4f*)&rb[4] = *(const v4f*)(src + 4);
        } else {                      // B[N][K]: 8 floats along k, fixed row n
            const int n  = tid & 127;
            const int kq = (tid >> 7) * 8;
            const float* src = B + (size_t)(bN + n) * K + k0 + kq;
            *(v4f*)&rb[0] = *(const v4f*)src;
            *(v4f*)&rb[4] = *(const v4f*)(src + 4);
        }
    };
    // pair-interleaved store: BsP[k/2][2n + (k&1)] = B[k][n]
    auto storeB = [&](int buf) {
        if (!TRANS_B) {
            const int r  = tid >> 4;
            const int cq = (tid & 15) * 8;
            const int p  = r >> 1, q = r & 1;
#pragma unroll
            for (int i = 0; i < 8; ++i)
                BsP[buf][p][2 * (cq + i) + q] = rb[i];
        } else {
            const int n  = tid & 127;
            const int kq = (tid >> 7) * 8;
#pragma unroll
            for (int i = 0; i < 8; ++i)
                BsP[buf][(kq + i) >> 1][2 * n + (i & 1)] = rb[i];
        }
    };

    auto compute = [&](int buf) {
#pragma unroll
        for (int kk = 0; kk < 16; kk += 4) {
            v2f af[2];
#pragma unroll
            for (int i = 0; i < 2; ++i)
                af[i] = *(const v2f*)&As[buf][wm + i * 16 + l16][kk + lh * 2];
            v2f bf[4];
#pragma unroll
            for (int j = 0; j < 4; ++j)
                bf[j] = *(const v2f*)&BsP[buf][(kk >> 1) + lh]
                                          [2 * (wn + j * 16 + l16)];
#pragma unroll
            for (int i = 0; i < 2; ++i)
#pragma unroll
                for (int j = 0; j < 4; ++j)
                    acc[i][j] = __builtin_amdgcn_wmma_f32_16x16x4_f32(
                        false, af[i], false, bf[j], (short)0, acc[i][j],
                        false, false);
        }
    };

    // ---------------- prologue: stage 0 ----------------
    issueA(0, 0);
    loadB(0);
    commitA(0);
    storeB(0);
    waitA();
    __syncthreads();

    // ---------------- main loop, double buffered ----------------
    int cur = 0;
    for (int k0 = 0; k0 < K; k0 += 16) {
        const bool more = (k0 + 16) < K;
        if (more) { issueA(k0 + 16, cur ^ 1); loadB(k0 + 16); }
        compute(cur);
        if (more) { commitA(cur ^ 1); storeB(cur ^ 1); }
        waitA();
        __syncthreads();
        cur ^= 1;
    }

    // ---- epilogue + store; C/D layout: VGPR r -> M = r + 8*lh, N = l16 ----
#pragma unroll
    for (int j = 0; j < 4; ++j) {
        const int col = bN + wn + j * 16 + l16;
        const float bv = (EPI == 2) ? bias[col] : 0.f;
#pragma unroll
        for (int i = 0; i < 2; ++i) {
#pragma unroll
            for (int r = 0; r < 8; ++r) {
                const int row = bM + wm + i * 16 + lh * 8 + r;
                float v = acc[i][j][r];
                if (EPI == 1)      v = (v > 0.f) ? v : (__expf(v) - 1.f); // ELU
                else if (EPI == 2) { v += bv; v = v > 0.f ? v : 0.f; }    // bias+ReLU
                C[(size_t)row * ldc + coff + col] = v;
            }
        }
    }
}

// ---------------------------------------------------------------------------
// Per-row sum of squares
// ---------------------------------------------------------------------------
__global__ void row_sumsq(const float* __restrict__ X, float* __restrict__ sq, int D)
{
    __shared__ float red[256];
    const int i = blockIdx.x, tid = threadIdx.x;
    float s = 0.f;
    for (int j = tid; j < D; j += 256) { float v = X[(size_t)i * D + j]; s += v * v; }
    red[tid] = s; __syncthreads();
    for (int o = 128; o > 0; o >>= 1) {
        if (tid < o) red[tid] += red[tid + o];
        __syncthreads();
    }
    if (tid == 0) sq[i] = red[0];
}

__global__ void normalize_rows(const float* __restrict__ X, const float* __restrict__ sq,
                               float* __restrict__ Xn, int D)
{
    const int t = blockIdx.x * blockDim.x + threadIdx.x;
    const int i = t / D;
    Xn[t] = X[t] * rsqrtf(sq[i]);
}

// ---------------------------------------------------------------------------
// Row top-(K+1) -> adjacency entries.
// mode 0 (euclid): score_j = 2*G_ij - sq_j  (== argmin d^2); entries = 1.0
// mode 1 (cosine): score_j = sim_ij; entries = sim  (0.0 if j==i, both modes)
// ---------------------------------------------------------------------------
__global__ void topk_adj(const float* __restrict__ S, const float* __restrict__ sq,
                         float* __restrict__ adj, int n, int mode)
{
    __shared__ float row[NN_];
    __shared__ float rmax[256];
    __shared__ int   ridx[256];
    const int i = blockIdx.x, tid = threadIdx.x;
    for (int j = tid; j < n; j += 256) {
        const float s = S[(size_t)i * n + j];
        row[j] = (mode == 0) ? (2.f * s - sq[j]) : s;
    }
    __syncthreads();
    for (int p = 0; p < KSEL; ++p) {
        float best = -3.4e38f; int bi = 0;
        for (int j = tid; j < n; j += 256) {
            const float v = row[j];
            if (v > best) { best = v; bi = j; }
        }
        rmax[tid] = best; ridx[tid] = bi;
        __syncthreads();
        for (int o = 128; o > 0; o >>= 1) {
            if (tid < o && rmax[tid + o] > rmax[tid]) {
                rmax[tid] = rmax[tid + o]; ridx[tid] = ridx[tid + o];
            }
            __syncthreads();
        }
        if (tid == 0) {
            const int j = ridx[0];
            float val;
            if (mode == 0) val = (j == i) ? 0.f : 1.f;
            else           val = (j == i) ? 0.f : rmax[0];
            adj[(size_t)i * n + j] = val;
            row[j] = -3.4e38f;
        }
        __syncthreads();
    }
}

__global__ void symmetrize_max(float* __restrict__ adj, int n)
{
    const int t = blockIdx.x * blockDim.x + threadIdx.x;
    const int i = t / n, j = t % n;
    if (i < j) {
        const float a = adj[(size_t)i * n + j];
        const float b = adj[(size_t)j * n + i];
        const float m = fmaxf(a, b);
        adj[(size_t)i * n + j] = m;
        adj[(size_t)j * n + i] = m;
    }
}

// ---------------------------------------------------------------------------
// f1[i] = Wh[i,:] . a[0:F] ; f2[i] = Wh[i,:] . a[F:2F]   (one block per row)
// ---------------------------------------------------------------------------
__global__ void matvec2(const float* __restrict__ Wh, const float* __restrict__ a,
                        float* __restrict__ f1, float* __restrict__ f2, int F)
{
    __shared__ float r1[256], r2[256];
    const int i = blockIdx.x, tid = threadIdx.x;
    float s1 = 0.f, s2 = 0.f;
    for (int j = tid; j < F; j += 256) {
        const float w = Wh[(size_t)i * F + j];
        s1 += w * a[j];
        s2 += w * a[F + j];
    }
    r1[tid] = s1; r2[tid] = s2; __syncthreads();
    for (int o = 128; o > 0; o >>= 1) {
        if (tid < o) { r1[tid] += r1[tid + o]; r2[tid] += r2[tid + o]; }
        __syncthreads();
    }
    if (tid == 0) { f1[i] = r1[0]; f2[i] = r2[0]; }
}

// ---------------------------------------------------------------------------
// att[i,:] = softmax_j( adj_ij>0 ? leaky_relu(f1_i + f2_j) : NEG )
// ---------------------------------------------------------------------------
__global__ void att_softmax(const float* __restrict__ adj, const float* __restrict__ f1,
                            const float* __restrict__ f2, float* __restrict__ att, int n)
{
    __shared__ float e[NN_];
    __shared__ float red[256];
    const int i = blockIdx.x, tid = threadIdx.x;
    const float fi = f1[i];
    float lmax = -3.4e38f;
    for (int j = tid; j < n; j += 256) {
        float x = fi + f2[j];
        x = (x > 0.f) ? x : ALPHA_ * x;                    // leaky relu
        x = (adj[(size_t)i * n + j] > 0.f) ? x : NEG_;     // mask non-edges
        e[j] = x;
        lmax = fmaxf(lmax, x);
    }
    red[tid] = lmax; __syncthreads();
    for (int o = 128; o > 0; o >>= 1) {
        if (tid < o) red[tid] = fmaxf(red[tid], red[tid + o]);
        __syncthreads();
    }
    const float m = red[0]; __syncthreads();
    float lsum = 0.f;
    for (int j = tid; j < n; j += 256) {
        const float v = __expf(e[j] - m);   // NEG rows underflow to exact 0
        e[j] = v; lsum += v;
    }
    red[tid] = lsum; __syncthreads();
    for (int o = 128; o > 0; o >>= 1) {
        if (tid < o) red[tid] += red[tid + o];
        __syncthreads();
    }
    const float inv = 1.f / red[0];
    for (int j = tid; j < n; j += 256)
        att[(size_t)i * n + j] = e[j] * inv;
}

// ---------------------------------------------------------------------------
// Final MLP layer: out[i] = X[i,:] . w + b   (one wave32 per row)
// ---------------------------------------------------------------------------
__global__ void matvec_bias(const float* __restrict__ X, const float* __restrict__ w,
                            const float* __restrict__ b, float* __restrict__ out, int K)
{
    const int row  = blockIdx.x * 8 + (threadIdx.x >> 5);
    const int lane = threadIdx.x & 31;
    float s = 0.f;
    for (int j = lane; j < K; j += 32) s += X[(size_t)row * K + j] * w[j];
    for (int o = 16; o > 0; o >>= 1) s += __shfl_down(s, o, 32);
    if (lane == 0) out[row] = s + b[0];
}

// ---------------------------------------------------------------------------
// Host-side helpers
// ---------------------------------------------------------------------------
static void gemm(hipStream_t s, const float* A, const float* B, const float* bias,
                 float* C, int M, int N, int K, int ldc, int coff,
                 bool transB, int epi)
{
    dim3 grid(N / 128, M / 128), block(256);
    if (!transB) {
        if (epi == 0)      gemm_wmma_f32<false, 0><<<grid, block, 0, s>>>(A, B, bias, C, M, N, K, ldc, coff);
        else if (epi == 1) gemm_wmma_f32<false, 1><<<grid, block, 0, s>>>(A, B, bias, C, M, N, K, ldc, coff);
        else               gemm_wmma_f32<false, 2><<<grid, block, 0, s>>>(A, B, bias, C, M, N, K, ldc, coff);
    } else {
        if (epi == 0)      gemm_wmma_f32<true, 0><<<grid, block, 0, s>>>(A, B, bias, C, M, N, K, ldc, coff);
        else if (epi == 1) gemm_wmma_f32<true, 1><<<grid, block, 0, s>>>(A, B, bias, C, M, N, K, ldc, coff);
        else               gemm_wmma_f32<true, 2><<<grid, block, 0, s>>>(A, B, bias, C, M, N, K, ldc, coff);
    }
}

static void run_gat(hipStream_t s, const float* X, int D,
                    const float* W_h, const float* a_h,
                    const float* W_o, const float* a_o,
                    const float* adj, float* att, float* Wh, float* hid,
                    float* f1, float* f2, float* outbuf, int out_ldc, int out_off)
{
    const int n = NN_;
    for (int h = 0; h < 4; ++h) {
        gemm(s, X, W_h + (size_t)h * D * 256, nullptr, Wh, n, 256, D, 256, 0, false, 0);
        matvec2<<<n, 256, 0, s>>>(Wh, a_h + (size_t)h * 512, f1, f2, 256);
        att_softmax<<<n, 256, 0, s>>>(adj, f1, f2, att, n);
        gemm(s, att, Wh, nullptr, hid, n, 256, n, 1024, h * 256, false, 1); // elu
    }
    gemm(s, hid, W_o, nullptr, Wh, n, 256, 1024, 256, 0, false, 0);
    matvec2<<<n, 256, 0, s>>>(Wh, a_o, f1, f2, 256);
    att_softmax<<<n, 256, 0, s>>>(adj, f1, f2, att, n);
    gemm(s, att, Wh, nullptr, outbuf, n, 256, n, out_ldc, out_off, false, 1); // elu
}

extern "C" void kernel_launch(void* const* d_in, const int* in_sizes, int n_in,
                              void* d_out, int out_size, void* d_ws, size_t ws_size,
                              hipStream_t stream)
{
    const int n = NN_, D = 512;
    const float* user = (const float*)d_in[0];
    const float* food = (const float*)d_in[1];
    const float* uWh  = (const float*)d_in[2];
    const float* uah  = (const float*)d_in[3];
    const float* uWo  = (const float*)d_in[4];
    const float* uao  = (const float*)d_in[5];
    const float* fWh  = (const float*)d_in[6];
    const float* fah  = (const float*)d_in[7];
    const float* fWo  = (const float*)d_in[8];
    const float* fao  = (const float*)d_in[9];
    const float* W0 = (const float*)d_in[10]; const float* b0 = (const float*)d_in[11];
    const float* W1 = (const float*)d_in[12]; const float* b1 = (const float*)d_in[13];
    const float* W2 = (const float*)d_in[14]; const float* b2 = (const float*)d_in[15];
    const float* W3 = (const float*)d_in[16]; const float* b3 = (const float*)d_in[17];
    float* out = (float*)d_out;

    float* ws = (float*)d_ws;
    size_t off = 0;
    auto carve = [&](size_t cnt) { float* p = ws + off; off += cnt; return p; };
    float* S    = carve((size_t)n * n);
    float* adj  = carve((size_t)n * n);
    float* att  = carve((size_t)n * n);
    float* Xn   = carve((size_t)n * D);
    float* sq   = carve(n);
    float* f1   = carve(n);
    float* f2   = carve(n);
    float* Wh   = carve((size_t)n * 256);
    float* hid  = carve((size_t)n * 1024);
    float* pr   = carve((size_t)n * 512);
    float* t1   = carve((size_t)n * 512);
    float* t2   = carve((size_t)n * 256);
    float* t3   = carve((size_t)n * 128);
    (void)ws_size; (void)in_sizes; (void)n_in; (void)out_size;

    // ---------------- USER modality: euclid kNN adjacency ----------------
    row_sumsq<<<n, 256, 0, stream>>>(user, sq, D);
    gemm(stream, user, user, nullptr, S, n, n, D, n, 0, true, 0);   // G = X X^T
    (void)hipMemsetAsync(adj, 0, (size_t)n * n * sizeof(float), stream);
    topk_adj<<<n, 256, 0, stream>>>(S, sq, adj, n, 0);
    symmetrize_max<<<(n * n) / 256, 256, 0, stream>>>(adj, n);
    run_gat(stream, user, D, uWh, uah, uWo, uao, adj, att, Wh, hid, f1, f2,
            pr, 512, 0);

    // ---------------- FOOD modality: cosine kNN adjacency ----------------
    row_sumsq<<<n, 256, 0, stream>>>(food, sq, D);
    normalize_rows<<<(n * D) / 256, 256, 0, stream>>>(food, sq, Xn, D);
    gemm(stream, Xn, Xn, nullptr, S, n, n, D, n, 0, true, 0);       // sim = Xn Xn^T
    (void)hipMemsetAsync(adj, 0, (size_t)n * n * sizeof(float), stream);
    topk_adj<<<n, 256, 0, stream>>>(S, sq, adj, n, 1);
    symmetrize_max<<<(n * n) / 256, 256, 0, stream>>>(adj, n);
    run_gat(stream, food, D, fWh, fah, fWo, fao, adj, att, Wh, hid, f1, f2,
            pr, 512, 256);

    // ---------------- MLP head ----------------
    gemm(stream, pr, W0, b0, t1, n, 512, 512, 512, 0, false, 2);
    gemm(stream, t1, W1, b1, t2, n, 256, 512, 256, 0, false, 2);
    gemm(stream, t2, W2, b2, t3, n, 128, 256, 128, 0, false, 2);
    matvec_bias<<<n / 8, 256, 0, stream>>>(t3, W3, b3, out, 128);   // [N,1]
}